// DiscreteHawkes_36782099923577
// MI455X (gfx1250) — compile-verified
//
#include <hip/hip_runtime.h>
#include <hip/hip_bf16.h>

// Problem constants (from the reference)
#define T_DIM 8192
#define S_DIM 1024
#define B_DIM 8192
#define CHUNK 256
#define NCHUNK (T_DIM / CHUNK)   // 32

typedef __attribute__((ext_vector_type(2))) float v2f;
typedef __attribute__((ext_vector_type(8))) float v8f;

// -------- Phase 1: local decayed scan of obs columns -------------------------
// G[t,sp] (local, zero carry-in) for t in chunk k; gend[k,sp] = value one past
// chunk end. Thread = (chunk k, column sp); lanes span consecutive sp =>
// fully coalesced loads/stores at each t.
__global__ void hawkes_scan_local(const int* __restrict__ obs,
                                  const float* __restrict__ beta,
                                  float* __restrict__ G,
                                  float* __restrict__ gend) {
    int gid = blockIdx.x * blockDim.x + threadIdx.x;   // 0 .. NCHUNK*S-1
    int k   = gid >> 10;                               // / S_DIM
    int sp  = gid & (S_DIM - 1);                       // % S_DIM
    float a = expf(-beta[0]);
    int t0  = k * CHUNK;
    const int* op = obs + (size_t)t0 * S_DIM + sp;
    float*     gp = G   + (size_t)t0 * S_DIM + sp;
    float g = 0.0f;
    for (int i = 0; i < CHUNK; ++i) {
        gp[(size_t)i * S_DIM] = g;
        g = a * (g + (float)op[(size_t)i * S_DIM]);
    }
    gend[k * S_DIM + sp] = g;
}

// -------- Phase 2: serial carry combine across chunks (per column) -----------
__global__ void hawkes_carry(const float* __restrict__ gend,
                             const float* __restrict__ beta,
                             float* __restrict__ carry) {
    int sp = blockIdx.x * blockDim.x + threadIdx.x;    // 0 .. S-1
    float aC = expf(-beta[0] * (float)CHUNK);          // a^CHUNK
    float c = 0.0f;
    for (int k = 0; k < NCHUNK; ++k) {
        carry[k * S_DIM + sp] = c;
        c = gend[k * S_DIM + sp] + aC * c;
    }
}

// -------- Phase 3: add decayed carry into each chunk -------------------------
__global__ void hawkes_fixup(const float* __restrict__ carry,
                             const float* __restrict__ beta,
                             float* __restrict__ G) {
    int gid = blockIdx.x * blockDim.x + threadIdx.x;
    int k   = gid >> 10;
    int sp  = gid & (S_DIM - 1);
    float q = carry[k * S_DIM + sp];
    if (q == 0.0f) return;                             // chunk 0, or fully decayed
    float a = expf(-beta[0]);
    int t0  = k * CHUNK;
    float* gp = G + (size_t)t0 * S_DIM + sp;
    for (int i = 0; i < CHUNK; ++i) {
        gp[(size_t)i * S_DIM] += q;                    // q = a^i * carry
        q *= a;
    }
}

// -------- Phase 4: 16-query batched dot products via f32 WMMA ----------------
// One wave per group of 16 queries. A = 16 rows of G (16 x 1024),
// B = 16 columns of alpha (1024 x 16). Accumulate 16x16 f32 tile with
// V_WMMA_F32_16X16X4_F32 over K=1024, read the diagonal, apply relu/mu/beta.
__global__ void hawkes_query_wmma(const int* __restrict__ tq,
                                  const int* __restrict__ sq,
                                  const float* __restrict__ G,
                                  const float* __restrict__ alpha,
                                  const float* __restrict__ beta,
                                  const float* __restrict__ mu,
                                  float* __restrict__ out) {
    const int lane  = threadIdx.x & 31;
    const int wave  = threadIdx.x >> 5;
    const int group = blockIdx.x * (blockDim.x >> 5) + wave;   // 0 .. B/16-1

    const int m  = lane & 15;                  // row of A / col of B this lane feeds
    const int tm = tq[group * 16 + m];
    const int sn = sq[group * 16 + m];
    const int c0 = (lane < 16) ? 0 : 2;        // K-pair held by this half-wave

    const float* __restrict__ Grow = G + (size_t)tm * S_DIM;
    const float* __restrict__ Bcol = alpha + sn;

    v8f acc = {};
    #pragma unroll 4
    for (int k = 0; k < S_DIM; k += 4) {
        // A fragment: G[tm][k+c0], G[tm][k+c0+1] (8B-aligned contiguous pair)
        v2f av = *(const v2f*)(Grow + k + c0);
        // B fragment: alpha[k+c0][sn], alpha[k+c0+1][sn] (column-strided)
        v2f bv;
        bv.x = Bcol[(size_t)(k + c0) * S_DIM];
        bv.y = Bcol[(size_t)(k + c0 + 1) * S_DIM];
        acc = __builtin_amdgcn_wmma_f32_16x16x4_f32(
            /*neg_a=*/false, av, /*neg_b=*/false, bv,
            /*c_mod=*/(short)0, acc, /*reuse_a=*/false, /*reuse_b=*/false);
    }

    // Diagonal extraction:
    //   m in 0..7  -> VGPR m   at lane m       (lanes 0-15: M=r, N=lane)
    //   m in 8..15 -> VGPR m-8 at lane m+16    (lanes 16-31: M=r+8, N=lane-16)
    const int r = lane & 7;
    float v = acc[0];
    if (r == 1) v = acc[1];
    if (r == 2) v = acc[2];
    if (r == 3) v = acc[3];
    if (r == 4) v = acc[4];
    if (r == 5) v = acc[5];
    if (r == 6) v = acc[6];
    if (r == 7) v = acc[7];

    const bool active = (lane < 8) || (lane >= 24);
    if (active) {
        int mOut = (lane < 8) ? lane : (lane - 16);
        int b = group * 16 + mOut;             // sn/tm already match this m
        float lam = mu[sn] + beta[0] * v;
        out[b] = fmaxf(0.0f, lam);
    }
}

extern "C" void kernel_launch(void* const* d_in, const int* in_sizes, int n_in,
                              void* d_out, int out_size, void* d_ws, size_t ws_size,
                              hipStream_t stream) {
    const int*   t_idx = (const int*)d_in[0];     // [B]
    const int*   s_idx = (const int*)d_in[1];     // [B]
    const int*   obs   = (const int*)d_in[2];     // [T,S]
    const float* alpha = (const float*)d_in[3];   // [S,S]
    const float* beta  = (const float*)d_in[4];   // [1]
    const float* mu    = (const float*)d_in[5];   // [S]
    float*       out   = (float*)d_out;           // [B]

    // Workspace layout
    char* ws = (char*)d_ws;
    float* G     = (float*)ws;                                            // T*S f32 = 32 MiB
    float* gend  = (float*)(ws + (size_t)T_DIM * S_DIM * sizeof(float));  // NCHUNK*S f32
    float* carry = gend + (size_t)NCHUNK * S_DIM;                         // NCHUNK*S f32

    // Phase 1: local scans — NCHUNK*S threads
    hawkes_scan_local<<<(NCHUNK * S_DIM) / 256, 256, 0, stream>>>(obs, beta, G, gend);
    // Phase 2: carry combine — S threads
    hawkes_carry<<<S_DIM / 256, 256, 0, stream>>>(gend, beta, carry);
    // Phase 3: fixup — NCHUNK*S threads
    hawkes_fixup<<<(NCHUNK * S_DIM) / 256, 256, 0, stream>>>(carry, beta, G);
    // Phase 4: WMMA query dots — one wave per 16 queries, 8 waves/block
    hawkes_query_wmma<<<(B_DIM / 16) / 8, 256, 0, stream>>>(
        t_idx, s_idx, G, alpha, beta, mu, out);
}